// ExtractSplitPosition_41420664603030
// MI455X (gfx1250) — compile-verified
//
#include <hip/hip_runtime.h>
#include <cmath>

// ---------------------------------------------------------------------------
// CDNA5 feature detection (async global->LDS data mover path, gfx1250)
// ---------------------------------------------------------------------------
#ifndef __has_builtin
#define __has_builtin(x) 0
#endif

#if __has_builtin(__builtin_amdgcn_global_load_async_to_lds_b128)
#define HAVE_ASYNC_B128 1
#else
#define HAVE_ASYNC_B128 0
#endif

#if __has_builtin(__builtin_amdgcn_s_wait_asynccnt)
#define WAIT_ASYNC0() __builtin_amdgcn_s_wait_asynccnt(0)
#else
#define WAIT_ASYNC0() asm volatile("s_wait_asynccnt 0" ::: "memory")
#endif

// Builtin parameter type per hipcc diagnostic:
//   "__attribute__((__vector_size__(4 * sizeof(int)))) int __device__ *"
typedef int v4i_t __attribute__((vector_size(4 * sizeof(int))));
typedef __attribute__((address_space(1))) v4i_t* gv4i_p;   // global src
typedef __attribute__((address_space(3))) v4i_t* lv4i_p;   // LDS dst

#define BLOCK     256
#define MAXFW     4096
#define EPT       (MAXFW / BLOCK)   // 16 anchors per thread
#define MAX_OUT   50
#define FSTRIDE   16.0f
#define SCORE_TH  0.7f
#define DIST_TH   16.0f
#define NEG_BIG   (-1e30f)

__global__ __launch_bounds__(BLOCK)
void ExtractSplitPosition_nms_kernel(const float* __restrict__ logits,
                                     const float* __restrict__ delta,
                                     const float* __restrict__ realw,
                                     float* __restrict__ out_pos,   // [B,50,3]
                                     float* __restrict__ out_scr,   // [B,50,2]
                                     float* __restrict__ out_cls,   // [B,fw]
                                     int fw)
{
    __shared__ __attribute__((aligned(16))) float s_logit[MAXFW];      // raw logits; reused as onehot
    __shared__ __attribute__((aligned(16))) float s_delta[2 * MAXFW];  // raw deltas (persist through loop)
    __shared__ float s_pv[8];
    __shared__ int   s_pi[8];
    __shared__ float s_bc[2];            // [0]=picked center, [1]=valid flag
    __shared__ int   s_pickK[MAX_OUT];
    __shared__ float s_pickF[MAX_OUT];

    const int tid = threadIdx.x;
    const int b   = blockIdx.x;
    const float* gl = logits + (size_t)b * fw;
    const float* gd = delta  + (size_t)b * fw * 2;
    const float maxw = realw[b] - 1.0f;

    // -------- Phase A: stage row into LDS (async data-mover path) --------
    const int nchunkL = fw >> 2;       // float4 chunks of logits
    const int nchunkD = fw >> 1;       // float4 chunks of deltas (2*fw/4)
#if HAVE_ASYNC_B128
    for (int q = tid; q < nchunkL; q += BLOCK)
        __builtin_amdgcn_global_load_async_to_lds_b128(
            (gv4i_p)(gl + 4 * q), (lv4i_p)&s_logit[4 * q], 0, 0);
    for (int q = tid; q < nchunkD; q += BLOCK)
        __builtin_amdgcn_global_load_async_to_lds_b128(
            (gv4i_p)(gd + 4 * q), (lv4i_p)&s_delta[4 * q], 0, 0);
    WAIT_ASYNC0();
#else
    for (int q = tid; q < nchunkL; q += BLOCK)
        ((float4*)s_logit)[q] = ((const float4*)gl)[q];
    for (int q = tid; q < nchunkD; q += BLOCK)
        ((float4*)s_delta)[q] = ((const float4*)gd)[q];
#endif
    __syncthreads();

    // -------- Phase B: decode scores / centers into registers --------
    float sc[EPT], ce[EPT];
    unsigned alive = 0u;
#pragma unroll
    for (int j = 0; j < EPT; ++j) {
        const int i = j * BLOCK + tid;
        float s = NEG_BIG, c = 0.0f;
        if (i < fw) {
            const float x = s_logit[i];
            s = 1.0f / (1.0f + expf(-x));
            const float ic = ((float)i + 0.5f) * FSTRIDE;
            float p0 = s_delta[2 * i]     * FSTRIDE + ic;
            float p1 = s_delta[2 * i + 1] * FSTRIDE + ic;
            p0 = fminf(fmaxf(p0, 0.0f), maxw);
            p1 = fminf(fmaxf(p1, 0.0f), maxw);
            c = (p0 + p1) * 0.5f;
            if (s >= SCORE_TH) alive |= (1u << j);
        }
        sc[j] = s; ce[j] = c;
    }
    __syncthreads();   // everyone done reading s_logit raw (region reused later)

    // -------- Phase C: greedy NMS, 50 serial iterations --------
    for (int it = 0; it < MAX_OUT; ++it) {
        // per-thread masked argmax (first-max tie-break via (max v, min i))
        float bv = NEG_BIG; int bi = 0x7FFFFFFF;
#pragma unroll
        for (int j = 0; j < EPT; ++j) {
            const float v = ((alive >> j) & 1u) ? sc[j] : NEG_BIG;
            const int   i = j * BLOCK + tid;
            if (v > bv || (v == bv && i < bi)) { bv = v; bi = i; }
        }
        // wave32 xor-shuffle reduction
#pragma unroll
        for (int m = 16; m >= 1; m >>= 1) {
            const float ov = __shfl_xor(bv, m, 32);
            const int   oi = __shfl_xor(bi, m, 32);
            if (ov > bv || (ov == bv && oi < bi)) { bv = ov; bi = oi; }
        }
        if ((tid & 31) == 0) { s_pv[tid >> 5] = bv; s_pi[tid >> 5] = bi; }
        __syncthreads();

        if (tid == 0) {
            float fv = s_pv[0]; int fi = s_pi[0];
#pragma unroll
            for (int w = 1; w < 8; ++w) {
                const float ov = s_pv[w]; const int oi = s_pi[w];
                if (ov > fv || (ov == fv && oi < fi)) { fv = ov; fi = oi; }
            }
            const float f = (fv >= SCORE_TH) ? 1.0f : 0.0f;
            // recompute winner's clamped positions from staged deltas
            const float ic = ((float)fi + 0.5f) * FSTRIDE;
            float p0 = s_delta[2 * fi]     * FSTRIDE + ic;
            float p1 = s_delta[2 * fi + 1] * FSTRIDE + ic;
            p0 = fminf(fmaxf(p0, 0.0f), maxw);
            p1 = fminf(fmaxf(p1, 0.0f), maxw);
            const float c = (p0 + p1) * 0.5f;

            const size_t po = (size_t)b * MAX_OUT + it;
            out_pos[po * 3 + 0] = (f != 0.0f) ? p0 : 0.0f;
            out_pos[po * 3 + 1] = (f != 0.0f) ? p1 : 0.0f;
            out_pos[po * 3 + 2] = f;
            out_scr[po * 2 + 0] = (f != 0.0f) ? fv : 0.0f;
            out_scr[po * 2 + 1] = f;

            s_bc[0] = c; s_bc[1] = f;
            s_pickK[it] = (int)floorf(((f != 0.0f) ? c : 0.0f) / FSTRIDE);
            s_pickF[it] = f;
        }
        __syncthreads();

        const float ci = s_bc[0];
        if (s_bc[1] != 0.0f) {
            // keep only strictly-farther-than-threshold boxes (matches > in ref)
#pragma unroll
            for (int j = 0; j < EPT; ++j)
                if (fabsf(ce[j] - ci) <= DIST_TH) alive &= ~(1u << j);
        }
    }

    // -------- Phase D: one-hot cls_ids row (reuse s_logit region) --------
    __syncthreads();
    for (int q = tid; q < fw; q += BLOCK) s_logit[q] = 0.0f;
    __syncthreads();
    if (tid < MAX_OUT && s_pickF[tid] != 0.0f) {
        const int k = s_pickK[tid];
        if (k >= 0 && k < fw) s_logit[k] = 1.0f;   // drop-mode semantics
    }
    __syncthreads();
    float* oc = out_cls + (size_t)b * fw;
    for (int q = tid; q < (fw >> 2); q += BLOCK)
        ((float4*)oc)[q] = ((const float4*)s_logit)[q];
}

// ---------------------------------------------------------------------------
// Host launcher
// ---------------------------------------------------------------------------
extern "C" void kernel_launch(void* const* d_in, const int* in_sizes, int n_in,
                              void* d_out, int out_size, void* d_ws, size_t ws_size,
                              hipStream_t stream)
{
    (void)n_in; (void)out_size; (void)d_ws; (void)ws_size;

    const float* logits = (const float*)d_in[0];   // [B, fw]
    const float* delta  = (const float*)d_in[1];   // [B, fw, 2]
    // d_in[2] = img_width scalar (fw derived from sizes instead)
    const float* realw  = (const float*)d_in[3];   // [B]

    const int Bn = in_sizes[3];
    const int fw = in_sizes[0] / Bn;               // 4096 for the given config

    float* out     = (float*)d_out;
    float* out_pos = out;                                          // B*50*3
    float* out_scr = out + (size_t)Bn * MAX_OUT * 3;               // B*50*2
    float* out_cls = out + (size_t)Bn * MAX_OUT * 3
                         + (size_t)Bn * MAX_OUT * 2;               // B*fw

    ExtractSplitPosition_nms_kernel<<<dim3(Bn), dim3(BLOCK), 0, stream>>>(
        logits, delta, realw, out_pos, out_scr, out_cls, fw);
}